// SSIMLoss_63617055588412
// MI455X (gfx1250) — compile-verified
//
#include <hip/hip_runtime.h>

// SSIM loss (pytorch_msssim style, win=11, sigma=1.5, data_range=1) on
// 64x1x512x512 f32. Separable Gaussian convs mapped onto
// V_WMMA_F32_16X16X4_F32 via a banded weight matrix (K padded 26->28,
// 7 chunks of K=4).
//
// One wave32 computes one 16x32 tile of the 502x502 SSIM map (all 5 filtered
// channels: X, Y, X^2, Y^2, XY). 4 waves per block, per-wave LDS staging.
// Vertical pass: 3 N-halves x 7 K-chunks x 5 ch = 105 WMMA.
// Horizontal pass: 2 N-tiles x 7 K-chunks x 5 ch = 70 WMMA.
// Band constants come from a small LDS LUT (prevents cndmask-chain remat).

typedef __attribute__((ext_vector_type(2))) float v2f;
typedef __attribute__((ext_vector_type(4))) float v4f;
typedef __attribute__((ext_vector_type(8))) float v8f;

#define IMG_H 512
#define IMG_W 512
#define OUT_D 502              // 512 - 11 + 1
#define NIMG 64
#define WAVES_PER_BLK 4

#define TILE_R 16
#define TILE_C 32
#define TILES_Y 32             // ceil(502/16)
#define TILES_X 16             // ceil(502/32)
#define TILES_PER_IMG (TILES_Y * TILES_X)   // 512

#define IN_COLS 48             // TILE_C + 10 halo, padded to 3 N-halves
#define IN_ROWS 28             // TILE_R + 10 halo, padded to 7 K-chunks
#define IN_FLOATS (IN_COLS * IN_ROWS)       // 1344 per array
#define IN_GROUPS (IN_FLOATS / 4)           // 336 float4 groups per array
#define V_STRIDE 48
#define V_CH (TILE_R * V_STRIDE)            // 768 floats per channel map
#define LDS_PER_WAVE (5 * V_CH)             // 3840 floats (>= 2*IN_FLOATS)

// 11-tap Gaussian (sigma=1.5), normalized (compile-time table).
__device__ __forceinline__ float wtap(int i) {
    const float w[11] = {
        0.00102838f, 0.00759876f, 0.03600077f, 0.10936009f, 0.21300553f,
        0.26601172f,
        0.21300553f, 0.10936009f, 0.03600077f, 0.00759876f, 0.00102838f };
    float r = 0.0f;
#pragma unroll
    for (int j = 0; j < 11; ++j) r = (i == j) ? w[j] : r;
    return r;
}

__device__ __forceinline__ v8f wmma_f32(v2f a, v2f b, v8f c) {
    // D = A(16x4) x B(4x16) + C(16x16), full f32
    return __builtin_amdgcn_wmma_f32_16x16x4_f32(
        false, a, false, b, (short)0, c, false, false);
}

__global__ void __launch_bounds__(WAVES_PER_BLK * 32)
ssim_tile_kernel(const float* __restrict__ X, const float* __restrict__ Y,
                 double* __restrict__ accum) {
    __shared__ float lds_all[WAVES_PER_BLK * LDS_PER_WAVE];
    __shared__ float wlut[16];   // [0]=0, [1..11]=taps, [12..]=0

    const int lane = threadIdx.x & 31;
    const int wave = threadIdx.x >> 5;
    float* wl = &lds_all[wave * LDS_PER_WAVE];

    // Every wave writes the same LUT values (benign identical-value race);
    // per-wave DS ordering guarantees this wave's reads see its own writes.
    if (lane < 16) {
        float v = (lane >= 1 && lane <= 11) ? wtap(lane - 1) : 0.0f;
        wlut[lane] = v;
    }

    // tile -> (image, tile row, tile col)
    const int t   = blockIdx.x * WAVES_PER_BLK + wave;
    const int img = t >> 9;                  // 512 tiles per image
    const int tid = t & 511;
    const int tr  = (tid >> 4) * TILE_R;     // output-row origin (0..496)
    const int tc  = (tid & 15) * TILE_C;     // output-col origin (0..480)

    const float* Xi = X + (size_t)img * (IMG_H * IMG_W);
    const float* Yi = Y + (size_t)img * (IMG_H * IMG_W);

    // ---- stage clamped 48 cols x 28 rows of X and Y, column-major in LDS,
    //      via b128 global loads (16B groups; clamped dups only ever feed
    //      zero band weights or masked outputs) ----
    for (int f = lane; f < IN_GROUPS; f += 32) {
        const int k  = f / 12;               // row within patch (0..27)
        const int gx = f - k * 12;           // float4 group within row (0..11)
        int grow = tr + k;       if (grow > IMG_H - 1) grow = IMG_H - 1;
        int gcb  = tc + gx * 4;  if (gcb  > IMG_W - 4) gcb  = IMG_W - 4;
        const size_t g = (size_t)grow * IMG_W + gcb;
        v4f xv = *(const v4f*)(Xi + g);
        v4f yv = *(const v4f*)(Yi + g);
        const int cb = gx * 4;
#pragma unroll
        for (int i = 0; i < 4; ++i) {
            wl[(cb + i) * IN_ROWS + k]             = xv[i];
            wl[IN_FLOATS + (cb + i) * IN_ROWS + k] = yv[i];
        }
    }

    // ---- banded Gaussian operand chunks from the LDS LUT (same register
    //      values serve as vertical A-operand and horizontal B-operand;
    //      lane&15 = M or N, lanes 16-31 at K+2) ----
    v2f wq[7];
    {
        const int half = lane >> 4;
        const int r    = lane & 15;
#pragma unroll
        for (int q = 0; q < 7; ++q) {
            int d  = 4 * q + 2 * half - r;       // band offset for .x
            int i0 = min(max(d + 1, 0), 12);
            int i1 = min(max(d + 2, 0), 12);
            wq[q].x = wlut[i0];
            wq[q].y = wlut[i1];
        }
    }

    // ---- vertical pass: V(16 out rows x 48 cols) for 5 channels ----
    v8f vx[3], vy[3], vxx[3], vyy[3], vxy[3];
#pragma unroll
    for (int h = 0; h < 3; ++h) {
        vx[h] = {}; vy[h] = {}; vxx[h] = {}; vyy[h] = {}; vxy[h] = {};
    }
    const int kb = 2 * (lane >> 4);
#pragma unroll
    for (int h = 0; h < 3; ++h) {
        const int c = (lane & 15) + 16 * h;  // B-operand column = patch col
#pragma unroll
        for (int q = 0; q < 7; ++q) {
            const int k = 4 * q + kb;
            v2f xb, yb;
            xb.x = wl[c * IN_ROWS + k];
            xb.y = wl[c * IN_ROWS + k + 1];
            yb.x = wl[IN_FLOATS + c * IN_ROWS + k];
            yb.y = wl[IN_FLOATS + c * IN_ROWS + k + 1];
            v2f xxb = xb * xb, yyb = yb * yb, xyb = xb * yb;
            vx[h]  = wmma_f32(wq[q], xb,  vx[h]);
            vy[h]  = wmma_f32(wq[q], yb,  vy[h]);
            vxx[h] = wmma_f32(wq[q], xxb, vxx[h]);
            vyy[h] = wmma_f32(wq[q], yyb, vyy[h]);
            vxy[h] = wmma_f32(wq[q], xyb, vxy[h]);
        }
    }

    // ---- spill V maps to LDS row-major (reuse input region; DS ops from one
    //      wave execute in order, so stores land after all vertical reads) ----
    float* V = wl;  // V[ch] at ch*768, 16 rows x 48 cols each
#pragma unroll
    for (int h = 0; h < 3; ++h) {
        const int col = (lane & 15) + 16 * h;
#pragma unroll
        for (int i = 0; i < 8; ++i) {
            const int row = i + 8 * (lane >> 4);
            V[0 * V_CH + row * V_STRIDE + col] = vx[h][i];
            V[1 * V_CH + row * V_STRIDE + col] = vy[h][i];
            V[2 * V_CH + row * V_STRIDE + col] = vxx[h][i];
            V[3 * V_CH + row * V_STRIDE + col] = vyy[h][i];
            V[4 * V_CH + row * V_STRIDE + col] = vxy[h][i];
        }
    }

    // ---- horizontal pass: 2 output N-tiles; A = V rows, B = wq band ----
    v8f mu1[2], mu2[2], fxx[2], fyy[2], fxy[2];
    {
        const int m = lane & 15;
#pragma unroll
        for (int nt = 0; nt < 2; ++nt) {
            mu1[nt] = {}; mu2[nt] = {}; fxx[nt] = {}; fyy[nt] = {}; fxy[nt] = {};
            const int base = m * V_STRIDE + 16 * nt;
#pragma unroll
            for (int q = 0; q < 7; ++q) {
                const int k = base + 4 * q + kb;
                v2f a;
                a.x = V[0 * V_CH + k]; a.y = V[0 * V_CH + k + 1];
                mu1[nt] = wmma_f32(a, wq[q], mu1[nt]);
                a.x = V[1 * V_CH + k]; a.y = V[1 * V_CH + k + 1];
                mu2[nt] = wmma_f32(a, wq[q], mu2[nt]);
                a.x = V[2 * V_CH + k]; a.y = V[2 * V_CH + k + 1];
                fxx[nt] = wmma_f32(a, wq[q], fxx[nt]);
                a.x = V[3 * V_CH + k]; a.y = V[3 * V_CH + k + 1];
                fyy[nt] = wmma_f32(a, wq[q], fyy[nt]);
                a.x = V[4 * V_CH + k]; a.y = V[4 * V_CH + k + 1];
                fxy[nt] = wmma_f32(a, wq[q], fxy[nt]);
            }
        }
    }

    // ---- SSIM map + boundary-masked accumulation ----
    // (a/b)*(c/d) fused as (a*c)/(b*d): one f32 divide per output.
    const float C1 = 1.0e-4f;   // (0.01*1)^2
    const float C2 = 9.0e-4f;   // (0.03*1)^2
    float s = 0.0f;
    const int rb = tr + 8 * (lane >> 4);
#pragma unroll
    for (int nt = 0; nt < 2; ++nt) {
        const int col = tc + 16 * nt + (lane & 15);
#pragma unroll
        for (int i = 0; i < 8; ++i) {
            float m1  = mu1[nt][i], m2 = mu2[nt][i];
            float s1  = fxx[nt][i] - m1 * m1;
            float s2  = fyy[nt][i] - m2 * m2;
            float s12 = fxy[nt][i] - m1 * m2;
            float num = (2.0f * m1 * m2 + C1) * (2.0f * s12 + C2);
            float den = (m1 * m1 + m2 * m2 + C1) * (s1 + s2 + C2);
            float ss  = num / den;
            bool ok   = (rb + i < OUT_D) && (col < OUT_D);
            s += ok ? ss : 0.0f;
        }
    }
#pragma unroll
    for (int off = 16; off > 0; off >>= 1) s += __shfl_down(s, off, 32);
    if (lane == 0) atomicAdd(accum, (double)s);
}

__global__ void ssim_zero_kernel(double* a) {
    if (threadIdx.x == 0 && blockIdx.x == 0) a[0] = 0.0;
}

__global__ void ssim_final_kernel(const double* __restrict__ a,
                                  float* __restrict__ out) {
    if (threadIdx.x == 0 && blockIdx.x == 0) {
        const double cnt = (double)NIMG * (double)OUT_D * (double)OUT_D;
        out[0] = 1.0f - (float)(a[0] / cnt);
    }
}

extern "C" void kernel_launch(void* const* d_in, const int* in_sizes, int n_in,
                              void* d_out, int out_size, void* d_ws, size_t ws_size,
                              hipStream_t stream) {
    const float* X = (const float*)d_in[0];
    const float* Y = (const float*)d_in[1];
    double* accum  = (double*)d_ws;

    ssim_zero_kernel<<<1, 32, 0, stream>>>(accum);

    const int total_tiles = NIMG * TILES_PER_IMG;            // 32768 waves
    const int blocks      = total_tiles / WAVES_PER_BLK;     // 8192 blocks
    ssim_tile_kernel<<<blocks, WAVES_PER_BLK * 32, 0, stream>>>(X, Y, accum);

    ssim_final_kernel<<<1, 32, 0, stream>>>(accum, (float*)d_out);
}